// RelativePosition_66520453481114
// MI455X (gfx1250) — compile-verified
//
#include <hip/hip_runtime.h>
#include <hip/hip_bf16.h>
#include <stdint.h>

// ---------------------------------------------------------------------------
// RelativePosition: out[b,i,j,:] = W[cls(b,i,j)] + bias
//   cls = (mask[b,i]&mask[b,j]) ? clip(ri[b,j]-ri[b,i],-bins,bins)+bins+1 : 0
// Pure streaming-store problem (1 GiB out @ 23.3 TB/s => ~46 us floor).
// Strategy: TDM tensor_load_to_lds stages W (C x D f32) in LDS once per block,
// then each lane gathers its fixed 16B chunk via ds_load_b128 and emits
// non-temporal global_store_b128.
// ---------------------------------------------------------------------------

typedef float        v4f __attribute__((ext_vector_type(4)));
typedef unsigned int v4u __attribute__((ext_vector_type(4)));
typedef int          v4i __attribute__((ext_vector_type(4)));
typedef int          v8i __attribute__((ext_vector_type(8)));

#define THREADS 256
#define MAX_WF  (80 * 128)   // max C*D floats staged in LDS (40 KB)
#define MAX_L   2048         // max sequence length for class cache (8 KB)

__global__ __launch_bounds__(THREADS) void relpos_gather_kernel(
    const int* __restrict__ ri,            // [B*L] residue_index (flat)
    const unsigned char* __restrict__ mk,  // [B*L] mask (numpy bool, 1 byte)
    const float* __restrict__ W,           // [C, D]
    const float* __restrict__ bias,        // [D]
    float* __restrict__ out,               // [B*L, L, D]
    int L, int C, int D, int bins)
{
    __shared__ __align__(16) float ldsW[MAX_WF];
    __shared__ int ldsCls[MAX_L];

    const int blk = blockIdx.x;        // flat (b*L + i)
    const int tid = threadIdx.x;
    const int b   = blk / L;

    // ---- Stage W into LDS with the Tensor Data Mover (wave 0 only) --------
    if (tid < 32) {
        const unsigned lds_off = (unsigned)(unsigned long long)(uintptr_t)&ldsW[0];
        const unsigned long long ga = (unsigned long long)(uintptr_t)W;

        // D# group 0: count=1 | lds_addr | global_addr(57b) | type=2
        v4u g0;
        g0.x = 1u;                                     // count=1, user mode
        g0.y = lds_off;                                // LDS byte address
        g0.z = (unsigned)ga;                           // global_addr[31:0]
        g0.w = (unsigned)((ga >> 32) & 0x01FFFFFFu)    // global_addr[56:32]
             | (2u << 30);                             // type = 2 ("image")

        // D# group 1: data_size=4B, tensor 128x66 (D x C), tile = full tensor,
        // row stride = D elements.
        v8i g1;
        g1.s0 = (int)(2u << 16);                                   // data_size=2 (4B)
        g1.s1 = (int)(((unsigned)D & 0xFFFFu) << 16);              // tensor_dim0 [15:0]
        g1.s2 = (int)((((unsigned)D >> 16) & 0xFFFFu)
              | (((unsigned)C & 0xFFFFu) << 16));                  // dim0 hi | dim1 lo
        g1.s3 = (int)((((unsigned)C >> 16) & 0xFFFFu)
              | (((unsigned)D & 0xFFFFu) << 16));                  // dim1 hi | tile_dim0
        g1.s4 = (int)((unsigned)C & 0xFFFFu);                      // tile_dim1 (tile_dim2=0)
        g1.s5 = D;                                                 // tensor_dim0_stride lo32
        g1.s6 = 0;
        g1.s7 = 0;

        v4i gz4 = {0, 0, 0, 0};                        // groups 2/3 unused (2D)
        v8i gz8 = {0, 0, 0, 0, 0, 0, 0, 0};            // extra group (clang-23 form)
        __builtin_amdgcn_tensor_load_to_lds(g0, g1, gz4, gz4, gz8, 0);
        __builtin_amdgcn_s_wait_tensorcnt(0);
    }

    // ---- Precompute class index per j into LDS ----------------------------
    const int ri_i = ri[blk];
    const int mi   = mk[blk];
    for (int j = tid; j < L; j += THREADS) {
        int rj = ri[b * L + j];
        int mj = mk[b * L + j];
        int d  = rj - ri_i;
        d = (d < -bins) ? -bins : d;
        d = (d >  bins) ?  bins : d;
        d = d + bins + 1;
        ldsCls[j] = (mi && mj) ? d : 0;
    }
    __syncthreads();

    // ---- Streaming gather: LDS row -> NT global_store_b128 ----------------
    const int cpr         = D >> 2;            // float4 chunks per row (32 for D=128)
    const int r           = tid / cpr;         // row slot within iteration
    const int c4          = (tid % cpr) * 4;   // fixed float offset in row
    const int rowsPerIter = THREADS / cpr;     // rows of output per iteration (8)

    const v4f bv = *(const v4f*)(bias + c4);   // bias chunk lives in VGPRs
    float* outBase = out + (size_t)blk * (size_t)L * (size_t)D + (size_t)c4;

    #pragma unroll 4
    for (int j = r; j < L; j += rowsPerIter) {
        const int cls = ldsCls[j];
        const v4f w = *(const v4f*)(&ldsW[cls * D + c4]);   // ds_load_b128
        v4f o = w + bv;
        __builtin_nontemporal_store(o, (v4f*)(outBase + (size_t)j * (size_t)D));
    }
}

extern "C" void kernel_launch(void* const* d_in, const int* in_sizes, int n_in,
                              void* d_out, int out_size, void* d_ws, size_t ws_size,
                              hipStream_t stream) {
    (void)n_in; (void)d_ws; (void)ws_size;
    const int*           ri = (const int*)d_in[0];
    const unsigned char* mk = (const unsigned char*)d_in[1];  // numpy bool = 1 byte
    const float*         W  = (const float*)d_in[2];
    const float*         bv = (const float*)d_in[3];
    float*              out = (float*)d_out;

    const int nri  = in_sizes[0];                 // B*L
    const int D    = in_sizes[3];                 // 128
    const int C    = in_sizes[2] / D;             // 66
    const int bins = (C - 2) / 2;                 // 32
    const long long L = (long long)out_size / ((long long)nri * (long long)D);

    dim3 grid((unsigned)nri), block(THREADS);
    hipLaunchKernelGGL(relpos_gather_kernel, grid, block, 0, stream,
                       ri, mk, W, bv, out, (int)L, C, D, bins);
}